// Dim0_StructuredSparse_35296041238594
// MI455X (gfx1250) — compile-verified
//
#include <hip/hip_runtime.h>

typedef __attribute__((ext_vector_type(2))) float v2f;
typedef __attribute__((ext_vector_type(8))) float v8f;

#define N_OUT    4096
#define N_IN     4096
#define N_SPARSE 1024
#define M_TOTAL  (4 * 2048)

#define BM 128
#define BN 128
#define BK 32
#define LK 33  // padded K-stride (odd -> conflict-free across 64 LDS banks)

__global__ __launch_bounds__(256)
void sparse_gather_gemm_wmma_f32(const float* __restrict__ x,
                                 const float* __restrict__ w,
                                 const int* __restrict__ idx,
                                 float* __restrict__ out)
{
    __shared__ int   idx_s[N_SPARSE];
    __shared__ float As[BM][LK];   // gathered x tile, [m][k]
    __shared__ float Bs[BN][LK];   // weight tile, n-major [n][k]

    const int tid  = threadIdx.x;
    const int lane = tid & 31;
    const int wave = tid >> 5;

    const int m0 = blockIdx.x * BM;   // 64 tiles
    const int n0 = blockIdx.y * BN;   // 32 tiles

    // wave grid inside block: 4 (M) x 2 (N); each wave owns 32 rows x 64 cols
    const int wm = wave & 3;
    const int wn = wave >> 2;

    // Preload all sparse column indices once.
    for (int i = tid; i < N_SPARSE; i += 256)
        idx_s[i] = idx[i];
    __syncthreads();

    v8f acc[2][4];
#pragma unroll
    for (int mi = 0; mi < 2; ++mi)
#pragma unroll
        for (int ni = 0; ni < 4; ++ni)
            acc[mi][ni] = (v8f)(0.0f);

    const int lhalf = lane & 15;
    const int hi    = lane >> 4;   // 0: lanes 0-15 (K 0,1), 1: lanes 16-31 (K 2,3)

    for (int k0 = 0; k0 < N_SPARSE; k0 += BK) {
        __syncthreads();  // previous compute finished before LDS overwrite

        // Gather A tile: 128x32 elements, 16 per thread.
        // One wave gathers one x row's 32 scattered columns (row = 16KB window).
#pragma unroll
        for (int i = 0; i < (BM * BK) / 256; ++i) {
            int e = tid + i * 256;
            int r = e >> 5;          // /BK
            int k = e & (BK - 1);    // %BK
            int col = idx_s[k0 + k];
            As[r][k] = x[(size_t)(m0 + r) * N_IN + col];
        }
        // Load B tile: w[n][k] contiguous in k -> coalesced 128B per wave.
#pragma unroll
        for (int i = 0; i < (BN * BK) / 256; ++i) {
            int e = tid + i * 256;
            int k = e & (BK - 1);
            int n = e >> 5;
            Bs[n][k] = w[(size_t)(n0 + n) * N_SPARSE + (k0 + k)];
        }
        // Prefetch next weight K-chunk (L2-resident, keep it hot near WGP).
        if (k0 + BK < N_SPARSE) {
            __builtin_prefetch(&w[(size_t)(n0 + (tid >> 1)) * N_SPARSE + (k0 + BK)], 0, 0);
        }
        __syncthreads();

#pragma unroll
        for (int kk = 0; kk < BK; kk += 4) {
            // A fragments (16x4 f32): lanes 0-15 hold K=kk,kk+1; lanes 16-31 K=kk+2,kk+3
            v2f afrag[2];
#pragma unroll
            for (int mi = 0; mi < 2; ++mi) {
                int row = wm * 32 + mi * 16 + lhalf;
                afrag[mi].x = As[row][kk + 2 * hi + 0];
                afrag[mi].y = As[row][kk + 2 * hi + 1];
            }
            // B fragments (4x16 f32): same K split across wave halves, N across lanes
            v2f bfrag[4];
#pragma unroll
            for (int ni = 0; ni < 4; ++ni) {
                int cn = wn * 64 + ni * 16 + lhalf;
                bfrag[ni].x = Bs[cn][kk + 2 * hi + 0];
                bfrag[ni].y = Bs[cn][kk + 2 * hi + 1];
            }
#pragma unroll
            for (int mi = 0; mi < 2; ++mi)
#pragma unroll
                for (int ni = 0; ni < 4; ++ni)
                    acc[mi][ni] = __builtin_amdgcn_wmma_f32_16x16x4_f32(
                        /*neg_a=*/false, afrag[mi],
                        /*neg_b=*/false, bfrag[ni],
                        /*c_mod=*/(short)0, acc[mi][ni],
                        /*reuse_a=*/false, /*reuse_b=*/false);
        }
    }

    // Store: VGPR r -> row m + (hi ? r+8 : r), col n + (lane&15); coalesced 64B runs.
#pragma unroll
    for (int mi = 0; mi < 2; ++mi) {
#pragma unroll
        for (int ni = 0; ni < 4; ++ni) {
            int col = n0 + wn * 64 + ni * 16 + lhalf;
#pragma unroll
            for (int r = 0; r < 8; ++r) {
                int row = m0 + wm * 32 + mi * 16 + hi * 8 + r;
                out[(size_t)row * N_OUT + col] = acc[mi][ni][r];
            }
        }
    }
}

extern "C" void kernel_launch(void* const* d_in, const int* in_sizes, int n_in,
                              void* d_out, int out_size, void* d_ws, size_t ws_size,
                              hipStream_t stream) {
    (void)in_sizes; (void)n_in; (void)out_size; (void)d_ws; (void)ws_size;
    const float* x   = (const float*)d_in[0];   // [4, 2048, 4096] f32
    const float* wgt = (const float*)d_in[1];   // [4096, 1024]    f32
    const int*   idx = (const int*)d_in[2];     // [1024] sorted column indices
    float*       out = (float*)d_out;           // [4, 2048, 4096] f32

    dim3 grid(M_TOTAL / BM, N_OUT / BN);        // 64 x 32
    dim3 block(256);                            // 8 wave32s
    sparse_gather_gemm_wmma_f32<<<grid, block, 0, stream>>>(x, wgt, idx, out);
}